// PatientGNN_24120536334551
// MI455X (gfx1250) — compile-verified
//
#include <hip/hip_runtime.h>
#include <hip/hip_bf16.h>

typedef __attribute__((ext_vector_type(2))) float v2f;
typedef __attribute__((ext_vector_type(8))) float v8f;

#define LDS_PITCH 132   // 128 + 4 pad -> LDS rows start 4 banks apart, conflict-free
#define HF 128          // hidden / feature width (fixed by problem)

// ---------------------------------------------------------------------------
// degree / normalization
// ---------------------------------------------------------------------------
__global__ void k_init_deg(float* __restrict__ deg, int n) {
    int i = blockIdx.x * blockDim.x + threadIdx.x;
    if (i < n) deg[i] = 1.0f;   // self-loop
}

__global__ void k_count_deg(const int* __restrict__ dst, float* __restrict__ deg, int E) {
    int e = blockIdx.x * blockDim.x + threadIdx.x;
    if (e < E) unsafeAtomicAdd(&deg[dst[e]], 1.0f);
}

__global__ void k_dinv(float* __restrict__ deg, int n) {
    int i = blockIdx.x * blockDim.x + threadIdx.x;
    if (i < n) {
        float d = deg[i];
        deg[i] = (d > 0.0f) ? rsqrtf(d) : 0.0f;   // deg >= 1 always, but match ref
    }
}

// ---------------------------------------------------------------------------
// WMMA fp32 GEMM: H[rows x 128] = Xin @ W (128x128), with optional fused BN
// applied to the input on load: xin = X*scale + shift.
// One block = 128 threads = 4 waves, computes a 16x128 output strip.
// ---------------------------------------------------------------------------
__global__ __launch_bounds__(128) void k_gemm128(
    const float* __restrict__ X, const float* __restrict__ W,
    float* __restrict__ H, const float* __restrict__ scale,
    const float* __restrict__ shift, int apply_bn)
{
    __shared__ float tile[16 * LDS_PITCH];
    const int rowbase = blockIdx.x * 16;
    const int t = threadIdx.x;

    // cooperative load of 16x128 input tile (coalesced), fuse BN scale/shift
    {
        const int row = t >> 3;          // 0..15
        const int c0  = (t & 7) * 16;    // 16 floats per thread
        const float* src = X + (size_t)(rowbase + row) * HF + c0;
        float* dl = tile + row * LDS_PITCH + c0;
#pragma unroll
        for (int j = 0; j < 16; j += 4) {
            float4 v = *(const float4*)(src + j);
            if (apply_bn) {
                v.x = v.x * scale[c0 + j + 0] + shift[c0 + j + 0];
                v.y = v.y * scale[c0 + j + 1] + shift[c0 + j + 1];
                v.z = v.z * scale[c0 + j + 2] + shift[c0 + j + 2];
                v.w = v.w * scale[c0 + j + 3] + shift[c0 + j + 3];
            }
            *(float4*)(dl + j) = v;
        }
    }
    __syncthreads();

    const int wave = t >> 5;        // 0..3 (wave32)
    const int lane = t & 31;
    const int hk   = lane >> 4;     // K-pair selector per ISA A-layout
    const int l16  = lane & 15;
    const int ct0  = wave * 32;     // two 16-wide col tiles per wave

    v8f acc0 = {};
    v8f acc1 = {};
    const float* lrow = tile + l16 * LDS_PITCH;

#pragma unroll 4
    for (int k = 0; k < HF; k += 4) {
        const int kk = k + 2 * hk;
        // A 16x4 f32 fragment: lane<16 -> (k,k+1), lane>=16 -> (k+2,k+3)
        v2f a = *(const v2f*)(lrow + kk);
        // B 4x16 f32 fragments (mirror layout), W row-major 128x128
        v2f b0, b1;
        b0.x = W[(size_t)kk * HF + ct0 + l16];
        b0.y = W[(size_t)(kk + 1) * HF + ct0 + l16];
        b1.x = W[(size_t)kk * HF + ct0 + 16 + l16];
        b1.y = W[(size_t)(kk + 1) * HF + ct0 + 16 + l16];
        acc0 = __builtin_amdgcn_wmma_f32_16x16x4_f32(false, a, false, b0,
                                                     (short)0, acc0, false, false);
        acc1 = __builtin_amdgcn_wmma_f32_16x16x4_f32(false, a, false, b1,
                                                     (short)0, acc1, false, false);
    }

    // D layout: VGPR r -> M = r + 8*hk, N = l16
    float* h0 = H + (size_t)(rowbase + hk * 8) * HF + ct0 + l16;
#pragma unroll
    for (int r = 0; r < 8; ++r) {
        h0[(size_t)r * HF]      = acc0[r];
        h0[(size_t)r * HF + 16] = acc1[r];
    }
}

// ---------------------------------------------------------------------------
// AGG[i] = H[i] * dinv[i]^2   (self-loop contribution, initializes AGG)
// one wave (32 lanes x float4) per node
// ---------------------------------------------------------------------------
__global__ void k_selfloop(const float* __restrict__ H, const float* __restrict__ dinv,
                           float* __restrict__ AGG, int n) {
    int idx = blockIdx.x * blockDim.x + threadIdx.x;
    int i = idx >> 5, lane = idx & 31;
    if (i >= n) return;
    float w = dinv[i]; w *= w;
    float4 v = *(const float4*)(H + (size_t)i * HF + lane * 4);
    v.x *= w; v.y *= w; v.z *= w; v.w *= w;
    *(float4*)(AGG + (size_t)i * HF + lane * 4) = v;
}

// ---------------------------------------------------------------------------
// scatter-add over edges: AGG[dst] += H[src] * dinv[src] * dinv[dst]
// one wave per edge; 4 native f32 atomics per lane
// ---------------------------------------------------------------------------
__global__ void k_scatter(const int* __restrict__ src, const int* __restrict__ dst,
                          const float* __restrict__ dinv, const float* __restrict__ H,
                          float* __restrict__ AGG, int E) {
    int gw = (blockIdx.x * blockDim.x + threadIdx.x) >> 5;
    int lane = threadIdx.x & 31;
    int nw = (gridDim.x * blockDim.x) >> 5;
    for (int e = gw; e < E; e += nw) {
        int s = src[e], d = dst[e];
        float nrm = dinv[s] * dinv[d];
        float4 hv = *(const float4*)(H + (size_t)s * HF + lane * 4);
        float* a = AGG + (size_t)d * HF + lane * 4;
        unsafeAtomicAdd(a + 0, hv.x * nrm);
        unsafeAtomicAdd(a + 1, hv.y * nrm);
        unsafeAtomicAdd(a + 2, hv.z * nrm);
        unsafeAtomicAdd(a + 3, hv.w * nrm);
    }
}

// ---------------------------------------------------------------------------
// BN statistics, fused with bias + relu (in place): Y = relu(AGG + b)
// thread = feature column; block covers a strip of rows (coalesced)
// ---------------------------------------------------------------------------
#define STRIP 256
__global__ __launch_bounds__(128) void k_stats(float* __restrict__ A, const float* __restrict__ b,
                                               float* __restrict__ cs, float* __restrict__ cs2,
                                               int n) {
    int f = threadIdx.x;                 // 0..127
    int r0 = blockIdx.x * STRIP;
    int r1 = r0 + STRIP; if (r1 > n) r1 = n;
    float bias = b[f];
    float s = 0.0f, s2 = 0.0f;
    for (int r = r0; r < r1; ++r) {
        float* p = A + (size_t)r * HF + f;
        float y = *p + bias;
        y = fmaxf(y, 0.0f);
        *p = y;
        s += y; s2 += y * y;
    }
    unsafeAtomicAdd(&cs[f], s);
    unsafeAtomicAdd(&cs2[f], s2);
}

__global__ void k_zero_stats(float* __restrict__ cs, float* __restrict__ cs2) {
    int t = threadIdx.x;
    if (t < HF) cs[t] = 0.0f; else cs2[t - HF] = 0.0f;
}

__global__ void k_bn_final(const float* __restrict__ cs, const float* __restrict__ cs2,
                           const float* __restrict__ g, const float* __restrict__ bt,
                           float* __restrict__ scv, float* __restrict__ shv, float invN) {
    int f = threadIdx.x;
    if (f < HF) {
        float mean = cs[f] * invN;
        float var  = cs2[f] * invN - mean * mean;
        float sc   = g[f] * rsqrtf(var + 1e-5f);
        scv[f] = sc;
        shv[f] = bt[f] - mean * sc;
    }
}

// ---------------------------------------------------------------------------
// classifier: out[i] = (Y[i]*scale + shift) @ Wc + bc   (128 -> 2)
// ---------------------------------------------------------------------------
__global__ void k_classifier(const float* __restrict__ Y, const float* __restrict__ scv,
                             const float* __restrict__ shv, const float* __restrict__ Wc,
                             const float* __restrict__ bc, float* __restrict__ out, int n) {
    int i = blockIdx.x * blockDim.x + threadIdx.x;
    if (i >= n) return;
    const float* y = Y + (size_t)i * HF;
    float a0 = bc[0], a1 = bc[1];
#pragma unroll 4
    for (int f = 0; f < HF; ++f) {
        float v = y[f] * scv[f] + shv[f];
        a0 += v * Wc[f * 2 + 0];
        a1 += v * Wc[f * 2 + 1];
    }
    out[(size_t)i * 2 + 0] = a0;
    out[(size_t)i * 2 + 1] = a1;
}

// ---------------------------------------------------------------------------
static inline int idiv_up(long long a, long long b) { return (int)((a + b - 1) / b); }

extern "C" void kernel_launch(void* const* d_in, const int* in_sizes, int n_in,
                              void* d_out, int out_size, void* d_ws, size_t ws_size,
                              hipStream_t stream) {
    const float* x  = (const float*)d_in[0];
    const int*   ei = (const int*)d_in[1];
    const int N = in_sizes[0] / HF;       // 100000
    const int E = in_sizes[1] / 2;        // 1600000
    const int* e_src = ei;
    const int* e_dst = ei + E;

    const float* Wl[3] = { (const float*)d_in[2], (const float*)d_in[6],  (const float*)d_in[10] };
    const float* bl[3] = { (const float*)d_in[3], (const float*)d_in[7],  (const float*)d_in[11] };
    const float* gl[3] = { (const float*)d_in[4], (const float*)d_in[8],  (const float*)d_in[12] };
    const float* tl[3] = { (const float*)d_in[5], (const float*)d_in[9],  (const float*)d_in[13] };
    const float* Wc = (const float*)d_in[14];
    const float* bc = (const float*)d_in[15];
    float* out = (float*)d_out;

    // workspace layout
    float* bufH = (float*)d_ws;                        // N*128  (GEMM output)
    float* bufA = bufH + (size_t)N * HF;               // N*128  (agg / relu'd activations)
    float* dinv = bufA + (size_t)N * HF;               // N      (deg -> deg^-1/2 in place)
    float* cs   = dinv + N;                            // 128
    float* cs2  = cs + HF;                             // 128
    float* scv  = cs2 + HF;                            // 128
    float* shv  = scv + HF;                            // 128

    // --- degree + normalization -------------------------------------------
    k_init_deg <<<idiv_up(N, 256), 256, 0, stream>>>(dinv, N);
    k_count_deg<<<idiv_up(E, 256), 256, 0, stream>>>(e_dst, dinv, E);
    k_dinv     <<<idiv_up(N, 256), 256, 0, stream>>>(dinv, N);

    // --- 3 GCN layers ------------------------------------------------------
    const float* xin = x;
    int apply_bn = 0;
    const int gemm_blocks    = N / 16;                          // 100000 % 16 == 0
    const int node_wave_blks = idiv_up((long long)N * 32, 256);
    const int edge_wave_blks = idiv_up((long long)E * 32, 256);

    for (int l = 0; l < 3; ++l) {
        k_gemm128 <<<gemm_blocks, 128, 0, stream>>>(xin, Wl[l], bufH, scv, shv, apply_bn);
        k_selfloop<<<node_wave_blks, 256, 0, stream>>>(bufH, dinv, bufA, N);
        k_scatter <<<edge_wave_blks, 256, 0, stream>>>(e_src, e_dst, dinv, bufH, bufA, E);
        k_zero_stats<<<1, 256, 0, stream>>>(cs, cs2);
        k_stats   <<<idiv_up(N, STRIP), 128, 0, stream>>>(bufA, bl[l], cs, cs2, N);
        k_bn_final<<<1, 128, 0, stream>>>(cs, cs2, gl[l], tl[l], scv, shv, 1.0f / (float)N);
        xin = bufA;
        apply_bn = 1;
    }

    // --- classifier --------------------------------------------------------
    k_classifier<<<idiv_up(N, 256), 256, 0, stream>>>(bufA, scv, shv, Wc, bc, out, N);
}